// LSTMStacked_51934744543240
// MI455X (gfx1250) — compile-verified
//
#include <hip/hip_runtime.h>

// ---------------------------------------------------------------------------
// Problem constants (match reference: B=3, T=262144, H=2, L=4)
// ---------------------------------------------------------------------------
#define T_LEN   262144      // 2^18
#define B_SZ    3
#define H_SZ    2
#define L_SZ    4
#define NROWS   (B_SZ * T_LEN)       // 786432 rows of the layer-0 projection GEMM
#define N_OUT   (B_SZ * T_LEN * H_SZ)

// LDS pipeline ring: 3 boundaries (w0->w1, w1->w2, w2->w3), 8 slots, 32 steps
#define CHUNK   32
#define SLOTS   8
#define NCHUNKS (T_LEN / CHUNK)

typedef __attribute__((ext_vector_type(2))) float v2f;
typedef __attribute__((ext_vector_type(8))) float v8f;

// ---------------------------------------------------------------------------
// Fast transcendentals on the CDNA5 TRANS pipe (v_exp_f32 is 2^x; v_rcp_f32)
// ---------------------------------------------------------------------------
__device__ __forceinline__ float fast_exp(float x) {
#if __has_builtin(__builtin_amdgcn_exp2f)
    return __builtin_amdgcn_exp2f(x * 1.4426950408889634f);
#else
    return __expf(x);
#endif
}
__device__ __forceinline__ float fast_rcp(float x) {
#if __has_builtin(__builtin_amdgcn_rcpf)
    return __builtin_amdgcn_rcpf(x);
#else
    return 1.0f / x;
#endif
}
__device__ __forceinline__ float sigm(float x) {
    return fast_rcp(1.0f + fast_exp(-x));
}
__device__ __forceinline__ float tanhv(float x) {
    // tanh(x) = 2*sigmoid(2x) - 1
    return __builtin_fmaf(2.0f, fast_rcp(1.0f + fast_exp(-2.0f * x)), -1.0f);
}
__device__ __forceinline__ void nap() {
#if __has_builtin(__builtin_amdgcn_s_sleep)
    __builtin_amdgcn_s_sleep(1);
#endif
}

// ---------------------------------------------------------------------------
// Kernel 1: layer-0 input projection via V_WMMA_F32_16X16X4_F32.
//   D[16x16] = A[16x4] * B[4x16] + C, per wave.
//   A rows  = 16 consecutive flat rows r = b*T + t of x (K = H = 2, padded to 4)
//   B cols  = 8 gate rows of W_ih[0] (padded to 16), B[k][n] = Wih[n][k]
//   C       = bias broadcast per column:  b_ih[0][n] + b_hh[0][n]
// Output layout proj[((t*3+b)*2 + j)*4 + g]  (g = gate i/f/g/o, j = hidden idx)
// so the scan's wave 0 reads two float4s per (t, b).
// A/B upper lane halves (K=2,3) are zero, so they contribute nothing.
// ---------------------------------------------------------------------------
__global__ __launch_bounds__(256) void proj0_wmma_kernel(
    const float* __restrict__ x,
    const float* __restrict__ W_ih,
    const float* __restrict__ b_ih,
    const float* __restrict__ b_hh,
    float* __restrict__ proj)
{
    const int wave = threadIdx.x >> 5;
    const int lane = threadIdx.x & 31;
    const int tile = blockIdx.x * 8 + wave;
    const int r0   = tile * 16;           // 16 GEMM rows per wave-tile
    const int nlo  = lane & 15;           // column index N (and row M offset sel)

    v2f a; v2f bm;
    if (lane < 16) {
        // A fragment: lanes 0..15 hold K=0,1 for row r0+lane
        a.x = x[(r0 + lane) * 2 + 0];
        a.y = x[(r0 + lane) * 2 + 1];
        // B fragment: lanes 0..15 hold K=0,1 for column n (gate row of W_ih)
        if (lane < 8) {
            bm.x = W_ih[lane * 2 + 0];
            bm.y = W_ih[lane * 2 + 1];
        } else {
            bm.x = 0.0f; bm.y = 0.0f;
        }
    } else {
        // K=2,3 padding — zeros
        a.x = 0.0f; a.y = 0.0f;
        bm.x = 0.0f; bm.y = 0.0f;
    }

    // C accumulator pre-loaded with combined bias (depends only on column N)
    float bias = (nlo < 8) ? (b_ih[nlo] + b_hh[nlo]) : 0.0f;
    v8f c;
#pragma unroll
    for (int r = 0; r < 8; ++r) c[r] = bias;

    v8f d = __builtin_amdgcn_wmma_f32_16x16x4_f32(
        false, a, false, bm, (short)0, c, false, false);

    // Scatter D into the scan-friendly layout
    if (nlo < 8) {
        const int j = nlo & 1;       // hidden index within gate
        const int g = nlo >> 1;      // gate: 0=i 1=f 2=g 3=o
        const int mbase = r0 + ((lane < 16) ? 0 : 8);
#pragma unroll
        for (int r = 0; r < 8; ++r) {
            const int row = mbase + r;           // flat row = b*T + t
            const int b   = row >> 18;           // T = 2^18
            const int t   = row & (T_LEN - 1);
            proj[(((t * 3 + b) * 2 + j) << 2) + g] = d[r];
        }
    }
}

// ---------------------------------------------------------------------------
// Kernel 2: 4-layer LSTM scan, layers pipelined across 4 waves of 1 workgroup.
//   wave w == layer w.  Lane b (b < 3) owns batch element b and keeps
//   h0,h1,c0,c1 plus all 8 gate weight rows in registers — no cross-lane ops
//   in the recurrence.  h chunks flow wave->wave through an LDS ring buffer
//   guarded by produced/consumed counters (volatile LDS + s_sleep polling).
// ---------------------------------------------------------------------------
__global__ __launch_bounds__(128) void lstm_scan_kernel(
    const float4* __restrict__ proj4,     // layer-0 projection (bias included)
    const float*  __restrict__ W_ih,
    const float*  __restrict__ W_hh,
    const float*  __restrict__ b_ih,
    const float*  __restrict__ b_hh,
    float* __restrict__ out)              // final layer h, layout [B][T][H]
{
    __shared__ float2   ring[3 * SLOTS * CHUNK * 3];   // [bnd][slot][step][b] = (h0,h1)
    __shared__ unsigned prod[3];
    __shared__ unsigned cons[3];

    const int w    = threadIdx.x >> 5;    // layer index
    const int lane = threadIdx.x & 31;
    if (threadIdx.x < 3) { prod[threadIdx.x] = 0u; cons[threadIdx.x] = 0u; }
    __syncthreads();

    // Per-lane (per batch element) constants, fully register resident.
    float wh0[8], wh1[8], wi0[8], wi1[8], bs[8];
#pragma unroll
    for (int n = 0; n < 8; ++n) { wh0[n]=wh1[n]=wi0[n]=wi1[n]=bs[n]=0.0f; }
    if (lane < 3) {
#pragma unroll
        for (int n = 0; n < 8; ++n) {
            wh0[n] = W_hh[w * 16 + n * 2 + 0];
            wh1[n] = W_hh[w * 16 + n * 2 + 1];
            if (w > 0) {
                wi0[n] = W_ih[w * 16 + n * 2 + 0];
                wi1[n] = W_ih[w * 16 + n * 2 + 1];
                bs[n]  = b_ih[w * 8 + n] + b_hh[w * 8 + n];
            }
        }
    }
    float h0 = 0.0f, h1 = 0.0f, c0 = 0.0f, c1 = 0.0f;

    volatile unsigned* vprod = prod;
    volatile unsigned* vcons = cons;
    const int b = lane;

    for (unsigned k = 0; k < NCHUNKS; ++k) {
        // --- flow control: consumer waits for data, producer waits for space ---
        if (w > 0) { while (vprod[w - 1] < k + 1) nap(); }
        if (w < 3) { while (k >= vcons[w] + SLOTS) nap(); }
        __threadfence_block();

        if (lane < 3) {
            const int slot = k & (SLOTS - 1);
            float2* rin  = &ring[(((w > 0 ? w - 1 : 0)) * SLOTS + slot) * CHUNK * 3];
            float2* rout = &ring[((w)                   * SLOTS + slot) * CHUNK * 3];
            const unsigned tbase = k * CHUNK;

#pragma unroll 4
            for (int s = 0; s < CHUNK; ++s) {
                const unsigned t = tbase + (unsigned)s;
                float pre[8];
                if (w == 0) {
                    const unsigned p = (t * 3u + (unsigned)b) * 2u;
                    float4 fa = proj4[p + 0];   // gates (i,f,g,o) for j=0
                    float4 fb = proj4[p + 1];   // gates (i,f,g,o) for j=1
                    pre[0] = fa.x; pre[2] = fa.y; pre[4] = fa.z; pre[6] = fa.w;
                    pre[1] = fb.x; pre[3] = fb.y; pre[5] = fb.z; pre[7] = fb.w;
                } else {
                    float2 hin = rin[s * 3 + b];
#pragma unroll
                    for (int n = 0; n < 8; ++n)
                        pre[n] = __builtin_fmaf(wi0[n], hin.x,
                                 __builtin_fmaf(wi1[n], hin.y, bs[n]));
                }
                // recurrent term  gates += h @ Whh^T
#pragma unroll
                for (int n = 0; n < 8; ++n)
                    pre[n] = __builtin_fmaf(wh0[n], h0,
                             __builtin_fmaf(wh1[n], h1, pre[n]));

                const float ai0 = sigm(pre[0]),  ai1 = sigm(pre[1]);
                const float af0 = sigm(pre[2]),  af1 = sigm(pre[3]);
                const float ag0 = tanhv(pre[4]), ag1 = tanhv(pre[5]);
                const float ao0 = sigm(pre[6]),  ao1 = sigm(pre[7]);

                c0 = __builtin_fmaf(af0, c0, ai0 * ag0);
                c1 = __builtin_fmaf(af1, c1, ai1 * ag1);
                h0 = ao0 * tanhv(c0);
                h1 = ao1 * tanhv(c1);

                if (w < 3) {
                    rout[s * 3 + b] = make_float2(h0, h1);
                } else {
                    ((float2*)out)[(unsigned)b * T_LEN + t] = make_float2(h0, h1);
                }
            }
        }

        // --- publish: data stores (DS in-order per wave) then counters ---
        __threadfence_block();
        if (lane == 0) {
            if (w < 3) vprod[w]     = k + 1;
            if (w > 0) vcons[w - 1] = k + 1;
        }
    }
}

// ---------------------------------------------------------------------------
// Kernel 3/4: deterministic two-stage MSE reduction (fixed-order trees).
// ---------------------------------------------------------------------------
__global__ __launch_bounds__(256) void mse_partial_kernel(
    const float* __restrict__ out, const float* __restrict__ lab,
    float* __restrict__ partial, int N)
{
    __shared__ float sm[256];
    float acc = 0.0f;
    for (int i = blockIdx.x * 256 + threadIdx.x; i < N; i += 256 * 256) {
        const float d = out[i] - lab[i];
        acc = __builtin_fmaf(d, d, acc);
    }
    sm[threadIdx.x] = acc;
    __syncthreads();
#pragma unroll
    for (int off = 128; off > 0; off >>= 1) {
        if (threadIdx.x < off) sm[threadIdx.x] += sm[threadIdx.x + off];
        __syncthreads();
    }
    if (threadIdx.x == 0) partial[blockIdx.x] = sm[0];
}

__global__ __launch_bounds__(256) void mse_final_kernel(
    const float* __restrict__ partial, float* __restrict__ loss, int N)
{
    __shared__ float sm[256];
    sm[threadIdx.x] = partial[threadIdx.x];
    __syncthreads();
#pragma unroll
    for (int off = 128; off > 0; off >>= 1) {
        if (threadIdx.x < off) sm[threadIdx.x] += sm[threadIdx.x + off];
        __syncthreads();
    }
    if (threadIdx.x == 0) loss[0] = sm[0] / (float)N;
}

// ---------------------------------------------------------------------------
// Launch: projection GEMM (WMMA) -> pipelined scan -> deterministic MSE.
// ---------------------------------------------------------------------------
extern "C" void kernel_launch(void* const* d_in, const int* in_sizes, int n_in,
                              void* d_out, int out_size, void* d_ws, size_t ws_size,
                              hipStream_t stream) {
    const float* x      = (const float*)d_in[0];   // [B,T,H]
    const float* labels = (const float*)d_in[1];   // [B,T,H]
    const float* W_ih   = (const float*)d_in[2];   // [L,4H,H]
    const float* W_hh   = (const float*)d_in[3];   // [L,4H,H]
    const float* b_ih   = (const float*)d_in[4];   // [L,4H]
    const float* b_hh   = (const float*)d_in[5];   // [L,4H]
    float* out = (float*)d_out;                    // [B,T,H] then loss scalar

    float* proj    = (float*)d_ws;                 // 24*T floats (~25 MB)
    float* partial = proj + (size_t)24 * T_LEN;    // 256 floats

    // 1) Layer-0 input projection + bias as a WMMA GEMM over 786432 rows.
    //    786432 rows / 16 per wave / 8 waves per block = 6144 blocks.
    proj0_wmma_kernel<<<NROWS / 16 / 8, 256, 0, stream>>>(x, W_ih, b_ih, b_hh, proj);

    // 2) All 4 LSTM layers, pipelined across 4 waves of a single workgroup.
    lstm_scan_kernel<<<1, 128, 0, stream>>>((const float4*)proj,
                                            W_ih, W_hh, b_ih, b_hh, out);

    // 3) MSE(out, labels) -> out[N_OUT], deterministic fixed-shape reduction.
    mse_partial_kernel<<<256, 256, 0, stream>>>(out, labels, partial, N_OUT);
    mse_final_kernel<<<1, 256, 0, stream>>>(partial, out + N_OUT, N_OUT);
}